// MambaResidualBlock_24833500905965
// MI455X (gfx1250) — compile-verified
//
#include <hip/hip_runtime.h>
#include <hip/hip_bf16.h>
#include <cstdint>

// ---------------------------------------------------------------------------
// Mamba residual block for MI455X (gfx1250, wave32, WMMA + async LDS copy).
// fp32 everywhere; GEMMs use V_WMMA_F32_16X16X4_F32 (reference numerics) with
// GLOBAL_LOAD_ASYNC_TO_LDS_B128 double-buffered staging (ASYNCcnt).
// Block tile 64(M) x 128(N); each wave: 64x32 strip = 8 accumulators.
// ---------------------------------------------------------------------------

typedef __attribute__((ext_vector_type(2))) float v2f;
typedef __attribute__((ext_vector_type(8))) float v8f;

#define D_MODEL 768
#define D_INNER 1536
#define D_STATE 16
#define D_CONV  4
#define DT_RANK 48
#define XDB_N   (DT_RANK + 2 * D_STATE)   // 80
#define BATCH   4
#define SEQLEN  1024
#define MROWS   (BATCH * SEQLEN)          // 4096

#define KC      16                         // K chunk per staged tile
#define LDSW    20                         // padded LDS row stride (floats)

__device__ __forceinline__ float silu_f(float v) {
    return v / (1.0f + __expf(-v));
}

// ---------------------------------------------------------------------------
// RMSNorm: one block per row of x [4096, 768].
// ---------------------------------------------------------------------------
__global__ __launch_bounds__(256)
void rmsnorm_kernel(const float* __restrict__ x, const float* __restrict__ rms_w,
                    float* __restrict__ xn) {
    const int row = blockIdx.x;
    const int tid = threadIdx.x;
    const int lane = tid & 31, wave = tid >> 5;
    __shared__ float red[8];
    __shared__ float scale_sh;

    float s = 0.0f;
    #pragma unroll
    for (int i = 0; i < 3; ++i) {
        float v = x[row * D_MODEL + tid + i * 256];
        s += v * v;
    }
    #pragma unroll
    for (int off = 16; off > 0; off >>= 1) s += __shfl_down(s, off, 32);
    if (lane == 0) red[wave] = s;
    __syncthreads();
    if (tid == 0) {
        float t = 0.0f;
        #pragma unroll
        for (int j = 0; j < 8; ++j) t += red[j];
        scale_sh = rsqrtf(t * (1.0f / (float)D_MODEL) + 1e-5f);
    }
    __syncthreads();
    const float sc = scale_sh;
    #pragma unroll
    for (int i = 0; i < 3; ++i) {
        int c = tid + i * 256;
        xn[row * D_MODEL + c] = x[row * D_MODEL + c] * sc * rms_w[c];
    }
}

// ---------------------------------------------------------------------------
// fp32 WMMA GEMM with async double-buffered LDS staging.
//   out[m,n] = sum_k A[m*lda+k] * W[n*K+k] (+ epilogue)
//   mode 0: plain store
//   mode 1: softplus(acc + aux[n])          (dt projection)
//   mode 2: acc + aux[m*N + n]              (residual add)
// Block: 128 threads = 4 waves; block tile 64(M) x 128(N); each wave owns a
// 64x32 strip (8 accumulators; A frags reused across two N tiles, B frags
// reused across four M tiles). K must be a multiple of 16; M mult of 64.
// Out-of-range N rows are clamped (junk stays in unstored columns).
// ---------------------------------------------------------------------------
__global__ __launch_bounds__(128)
void gemm_wmma_f32(const float* __restrict__ A, const float* __restrict__ W,
                   float* __restrict__ out, int N, int K, int lda,
                   int mode, const float* __restrict__ aux) {
    __shared__ __align__(16) float As[2][64 * LDSW];    // [m][k]
    __shared__ __align__(16) float Ws[2][128 * LDSW];   // [n][k]

    const int tid  = threadIdx.x;
    const int wave = tid >> 5;
    const int lane = tid & 31;
    const int row  = lane & 15;
    const int hi   = lane >> 4;

    const int m0 = blockIdx.y * 64;
    const int n0 = blockIdx.x * 128;

    auto issue = [&](int k0, int buf) {
        #pragma unroll
        for (int j = 0; j < 2; ++j) {   // A tile: 64x16, two B128 per thread.
            int i  = tid + j * 128;
            int r  = i >> 2;
            int c4 = (i & 3) * 4;
            unsigned lds = (unsigned)(uintptr_t)&As[buf][r * LDSW + c4];
            const float* g = A + (size_t)(m0 + r) * lda + k0 + c4;
            asm volatile("global_load_async_to_lds_b128 %0, %1, off"
                         :: "v"(lds), "v"(g) : "memory");
        }
        #pragma unroll
        for (int j = 0; j < 4; ++j) {   // W tile: 128x16, four B128 per thread.
            int i  = tid + j * 128;
            int r  = i >> 2;
            int c4 = (i & 3) * 4;
            int n  = n0 + r; if (n > N - 1) n = N - 1;   // clamp, no divergence
            unsigned lds = (unsigned)(uintptr_t)&Ws[buf][r * LDSW + c4];
            const float* g = W + (size_t)n * K + k0 + c4;
            asm volatile("global_load_async_to_lds_b128 %0, %1, off"
                         :: "v"(lds), "v"(g) : "memory");
        }
    };

    v8f acc[4][2] = {};   // [m-tile][n-tile]

    issue(0, 0);
    const int nch = K / KC;
    for (int i = 0; i < nch; ++i) {
        const int buf = i & 1;
        if (i + 1 < nch) {
            issue((i + 1) * KC, buf ^ 1);
            asm volatile("s_wait_asynccnt 0x6" ::: "memory");  // chunk i resident
        } else {
            asm volatile("s_wait_asynccnt 0x0" ::: "memory");
        }
        __syncthreads();

        #pragma unroll
        for (int kk = 0; kk < KC; kk += 4) {
            const int kb = kk + 2 * hi;
            v2f b0, b1;
            b0.x = Ws[buf][(wave * 32 + row) * LDSW + kb];
            b0.y = Ws[buf][(wave * 32 + row) * LDSW + kb + 1];
            b1.x = Ws[buf][(wave * 32 + 16 + row) * LDSW + kb];
            b1.y = Ws[buf][(wave * 32 + 16 + row) * LDSW + kb + 1];
            #pragma unroll
            for (int mt = 0; mt < 4; ++mt) {
                v2f a;
                a.x = As[buf][(mt * 16 + row) * LDSW + kb];
                a.y = As[buf][(mt * 16 + row) * LDSW + kb + 1];
                acc[mt][0] = __builtin_amdgcn_wmma_f32_16x16x4_f32(
                    false, a, false, b0, (short)0, acc[mt][0], false, false);
                acc[mt][1] = __builtin_amdgcn_wmma_f32_16x16x4_f32(
                    false, a, false, b1, (short)0, acc[mt][1], false, false);
            }
        }
        __syncthreads();   // buffer free for overwrite next iteration
    }

    // C/D layout: vgpr r -> m = r + 8*hi, n = lane%16.
    const int na = n0 + wave * 32 + row;
    const int nb = na + 16;
    #pragma unroll
    for (int mt = 0; mt < 4; ++mt) {
        #pragma unroll
        for (int r = 0; r < 8; ++r) {
            const int m = m0 + mt * 16 + r + 8 * hi;
            float v0 = acc[mt][0][r];
            float v1 = acc[mt][1][r];
            if (mode == 1) {
                if (na < N) { v0 += aux[na];
                    v0 = (v0 > 0.f) ? (v0 + log1pf(__expf(-v0))) : log1pf(__expf(v0)); }
                if (nb < N) { v1 += aux[nb];
                    v1 = (v1 > 0.f) ? (v1 + log1pf(__expf(-v1))) : log1pf(__expf(v1)); }
            } else if (mode == 2) {
                if (na < N) v0 += aux[(size_t)m * N + na];
                if (nb < N) v1 += aux[(size_t)m * N + nb];
            }
            if (na < N) out[(size_t)m * N + na] = v0;
            if (nb < N) out[(size_t)m * N + nb] = v1;
        }
    }
}

// ---------------------------------------------------------------------------
// Causal depthwise conv1d (kernel 4) + SiLU over u = xz[:, :D_INNER].
// ---------------------------------------------------------------------------
__global__ __launch_bounds__(256)
void conv_silu_kernel(const float* __restrict__ xz, const float* __restrict__ conv_w,
                      const float* __restrict__ conv_b, float* __restrict__ uc) {
    const int idx = blockIdx.x * 256 + threadIdx.x;       // flattened (b,l,d)
    const int d  = idx % D_INNER;
    const int bl = idx / D_INNER;                         // b*SEQLEN + l
    const int l  = bl % SEQLEN;

    float acc = conv_b[d];
    #pragma unroll
    for (int k = 0; k < D_CONV; ++k) {
        int ls = l + k - (D_CONV - 1);
        if (ls >= 0)
            acc += xz[(size_t)(bl + k - (D_CONV - 1)) * (2 * D_INNER) + d]
                   * conv_w[d * D_CONV + k];
    }
    uc[idx] = silu_f(acc);
}

// ---------------------------------------------------------------------------
// Selective scan. One thread per (b,d) channel; 16-state recurrence in
// registers; B_t/C_t staged in LDS each step (single 32-lane load).
// Reads dt from y_io and writes gated output back in-place.
// ---------------------------------------------------------------------------
__global__ __launch_bounds__(256)
void scan_kernel(float* __restrict__ y_io,                 // in: dt, out: y
                 const float* __restrict__ uc,
                 const float* __restrict__ xdb,
                 const float* __restrict__ A_log,
                 const float* __restrict__ D_param,
                 const float* __restrict__ xz) {           // for z gate
    const int tid = threadIdx.x;
    const int d = blockIdx.x * 256 + tid;
    const int b = blockIdx.y;

    float A[D_STATE];
    #pragma unroll
    for (int s = 0; s < D_STATE; ++s) A[s] = -__expf(A_log[d * D_STATE + s]);
    const float Dp = D_param[d];

    float h[D_STATE];
    #pragma unroll
    for (int s = 0; s < D_STATE; ++s) h[s] = 0.0f;

    __shared__ float BC[2 * D_STATE];   // [0..15]=B_t, [16..31]=C_t

    for (int l = 0; l < SEQLEN; ++l) {
        const int rowidx = b * SEQLEN + l;
        __syncthreads();
        if (tid < 2 * D_STATE)
            BC[tid] = xdb[(size_t)rowidx * XDB_N + DT_RANK + tid];
        __syncthreads();

        const float dtv = y_io[(size_t)rowidx * D_INNER + d];
        const float uv  = uc[(size_t)rowidx * D_INNER + d];
        const float du  = dtv * uv;
        float y = 0.0f;
        #pragma unroll
        for (int s = 0; s < D_STATE; ++s) {
            float dA = __expf(dtv * A[s]);
            h[s] = dA * h[s] + du * BC[s];
            y += h[s] * BC[D_STATE + s];
        }
        const float zv = xz[(size_t)rowidx * (2 * D_INNER) + D_INNER + d];
        y_io[(size_t)rowidx * D_INNER + d] = (y + uv * Dp) * silu_f(zv);
    }
}

// ---------------------------------------------------------------------------
// Launcher
// ---------------------------------------------------------------------------
extern "C" void kernel_launch(void* const* d_in, const int* in_sizes, int n_in,
                              void* d_out, int out_size, void* d_ws, size_t ws_size,
                              hipStream_t stream) {
    const float* x       = (const float*)d_in[0];
    const float* rms_w   = (const float*)d_in[1];
    const float* W_in    = (const float*)d_in[2];
    const float* conv_w  = (const float*)d_in[3];
    const float* conv_b  = (const float*)d_in[4];
    const float* W_xproj = (const float*)d_in[5];
    const float* W_dt    = (const float*)d_in[6];
    const float* b_dt    = (const float*)d_in[7];
    const float* A_log   = (const float*)d_in[8];
    const float* D_param = (const float*)d_in[9];
    const float* W_out   = (const float*)d_in[10];
    float* out = (float*)d_out;
    float* ws  = (float*)d_ws;

    // Workspace layout (fp32 elements)
    float* xn  = ws;                                        // 4096*768
    float* xz  = xn  + (size_t)MROWS * D_MODEL;             // 4096*3072
    float* uc  = xz  + (size_t)MROWS * 2 * D_INNER;         // 4096*1536
    float* xdb = uc  + (size_t)MROWS * D_INNER;             // 4096*80
    float* dty = xdb + (size_t)MROWS * XDB_N;               // 4096*1536 (dt, then y)

    // 1) RMSNorm
    rmsnorm_kernel<<<MROWS, 256, 0, stream>>>(x, rms_w, xn);

    // 2) Input projection: xz = xn @ W_in^T   [4096 x 3072], K=768
    gemm_wmma_f32<<<dim3((2 * D_INNER + 127) / 128, MROWS / 64), 128, 0, stream>>>(
        xn, W_in, xz, 2 * D_INNER, D_MODEL, D_MODEL, 0, nullptr);

    // 3) Causal depthwise conv + SiLU -> uc
    conv_silu_kernel<<<(MROWS * D_INNER) / 256, 256, 0, stream>>>(xz, conv_w, conv_b, uc);

    // 4) x-projection: xdb = uc @ W_xproj^T   [4096 x 80], K=1536
    gemm_wmma_f32<<<dim3((XDB_N + 127) / 128, MROWS / 64), 128, 0, stream>>>(
        uc, W_xproj, xdb, XDB_N, D_INNER, D_INNER, 0, nullptr);

    // 5) dt: softplus(dt_low @ W_dt^T + b_dt)  [4096 x 1536], K=48, lda=80
    gemm_wmma_f32<<<dim3((D_INNER + 127) / 128, MROWS / 64), 128, 0, stream>>>(
        xdb, W_dt, dty, D_INNER, DT_RANK, XDB_N, 1, b_dt);

    // 6) Selective scan (fuses +u*D and *silu(z)); writes y over dt buffer
    scan_kernel<<<dim3(D_INNER / 256, BATCH), 256, 0, stream>>>(
        dty, uc, xdb, A_log, D_param, xz);

    // 7) Output projection + residual: out = y @ W_out^T + x  [4096 x 768], K=1536
    gemm_wmma_f32<<<dim3((D_MODEL + 127) / 128, MROWS / 64), 128, 0, stream>>>(
        dty, W_out, out, D_MODEL, D_INNER, D_INNER, 2, x);
}